// GraphAutoencoder_49563922596249
// MI455X (gfx1250) — compile-verified
//
#include <hip/hip_runtime.h>
#include <hip/hip_bf16.h>

// ---------------------------------------------------------------------------
// GraphSAGE autoencoder for MI455X (gfx1250, wave32).
//   deg/neigh scatter -> mean -> fused dual-GEMM encoder (WMMA f32) -> decoder
// Workspace layout (floats): [ neigh_sum: N*256 ][ deg: N ]
// Output layout   (floats): [ reconstructed: N*256 ][ h: N*512 ]
// ---------------------------------------------------------------------------

#define N_NODES 50000
#define N_EDGES 800000
#define IN_DIM  256
#define HID_DIM 512
#define OUT_DIM 256

typedef float v2f __attribute__((ext_vector_type(2)));
typedef float v8f __attribute__((ext_vector_type(8)));

// ---------------------------------------------------------------------------
// Zero neigh_sum + deg (N*257 floats, divisible by 4 -> float4 stores).
// ---------------------------------------------------------------------------
__global__ void zero_ws_kernel(float4* __restrict__ p, int n4) {
    int i = blockIdx.x * blockDim.x + threadIdx.x;
    if (i < n4) p[i] = make_float4(0.f, 0.f, 0.f, 0.f);
}

// ---------------------------------------------------------------------------
// Edge scatter: 64 threads per edge, each owns a float4 slice of the 256-wide
// row. Gather of x[src] is fully coalesced per 64-thread group; scatter-adds
// land in L2 (neigh_sum = 51 MB < 192 MB L2).
// ---------------------------------------------------------------------------
__global__ void scatter_kernel(const float* __restrict__ x,
                               const int* __restrict__ src,
                               const int* __restrict__ dst,
                               float* __restrict__ neigh,
                               float* __restrict__ deg) {
    long long idx = (long long)blockIdx.x * blockDim.x + threadIdx.x;
    long long total = (long long)N_EDGES * 64;
    if (idx >= total) return;
    int edge  = (int)(idx >> 6);
    int chunk = (int)(idx & 63);
    int s = src[edge];
    int d = dst[edge];
    const float4* __restrict__ x4 = (const float4*)x;
    float4 v = x4[(long long)s * 64 + chunk];
    float* dstrow = neigh + (long long)d * IN_DIM + chunk * 4;
    atomicAdd(dstrow + 0, v.x);
    atomicAdd(dstrow + 1, v.y);
    atomicAdd(dstrow + 2, v.z);
    atomicAdd(dstrow + 3, v.w);
    if (chunk == 0) atomicAdd(deg + d, 1.0f);
}

// ---------------------------------------------------------------------------
// neigh_mean = neigh_sum / max(deg, 1)   (in place, float4 granularity)
// ---------------------------------------------------------------------------
__global__ void mean_kernel(float4* __restrict__ neigh,
                            const float* __restrict__ deg, int n4) {
    int i = blockIdx.x * blockDim.x + threadIdx.x;
    if (i >= n4) return;
    int row = i >> 6;                     // 64 float4 per 256-wide row
    float dg = deg[row];
    float inv = 1.0f / fmaxf(dg, 1.0f);
    float4 v = neigh[i];
    v.x *= inv; v.y *= inv; v.z *= inv; v.w *= inv;
    neigh[i] = v;
}

// ---------------------------------------------------------------------------
// WMMA fp32 GEMM: out = act( A1@W1 [+ A2@W2] + bias )
//   A: [M,K] row-major, W: [K,Ncols] row-major, out: [M,Ncols].
//   Block = 256 threads = 8 wave32 computing a 64x128 output tile.
//   Wave (wm = w&3, wn = w>>2) owns rows [wm*16,+16) x cols [wn*64,+64)
//   as FOUR 16x16 fp32 accumulators (v_wmma_f32_16x16x4_f32).
//   Both A and W tiles are stored K-contiguous per lane in LDS (stride 20),
//   so every fragment load is a single aligned ds_load_b64 (no v_mov glue).
// ---------------------------------------------------------------------------
#define LDS_K 20   // 16 + 4 pad floats; rows stay 16B aligned, b64 loads stay 8B aligned

template <bool DUAL, bool RELU>
__global__ void gemm_wmma_kernel(const float* __restrict__ A1,
                                 const float* __restrict__ W1,
                                 const float* __restrict__ A2,
                                 const float* __restrict__ W2,
                                 const float* __restrict__ bias,
                                 float* __restrict__ out,
                                 int M, int K, int Ncols) {
    __shared__ __align__(16) float xs [64  * LDS_K];   // A1 tile  [m][k]
    __shared__ __align__(16) float ns [64  * LDS_K];   // A2 tile  [m][k]
    __shared__ __align__(16) float w1t[128 * LDS_K];   // W1 tile  [n][k] (transposed)
    __shared__ __align__(16) float w2t[128 * LDS_K];   // W2 tile  [n][k] (transposed)

    const int tid  = threadIdx.x;
    const int lane = tid & 31;
    const int wave = tid >> 5;
    const int wm   = wave & 3;
    const int wn   = wave >> 2;
    const int m0   = blockIdx.x * 64;
    const int n0   = blockIdx.y * 128;

    v8f acc[4] = {{}, {}, {}, {}};

    const int half = lane >> 4;            // 0 for lanes 0-15, 1 for 16-31
    const int lidx = lane & 15;
    const int ml   = wm * 16 + lidx;       // local A row this lane reads
    const int cb   = wn * 64;              // wave's local column base

    for (int kk = 0; kk < K; kk += 16) {
        __syncthreads();
        // --- stage A tiles: 64 rows x 16 k, one float4 per thread per matrix
        {
            int row  = tid >> 2;           // 0..63
            int c    = tid & 3;            // float4 column 0..3
            int grow = m0 + row;
            float4 va = make_float4(0.f, 0.f, 0.f, 0.f);
            float4 vb = va;
            if (grow < M) {
                va = *(const float4*)&A1[(long long)grow * K + kk + c * 4];
                if (DUAL)
                    vb = *(const float4*)&A2[(long long)grow * K + kk + c * 4];
            }
            *(float4*)&xs[row * LDS_K + c * 4] = va;
            if (DUAL) *(float4*)&ns[row * LDS_K + c * 4] = vb;
        }
        // --- stage W tiles transposed: 16 k-rows x 128 cols -> [n][k] in LDS
#pragma unroll
        for (int it = 0; it < 2; ++it) {
            int idx = tid + it * 256;      // 0..511
            int row = idx >> 5;            // k-row 0..15
            int c   = idx & 31;            // float4 column 0..31
            float4 v = *(const float4*)&W1[(long long)(kk + row) * Ncols + n0 + c * 4];
            w1t[(c * 4 + 0) * LDS_K + row] = v.x;
            w1t[(c * 4 + 1) * LDS_K + row] = v.y;
            w1t[(c * 4 + 2) * LDS_K + row] = v.z;
            w1t[(c * 4 + 3) * LDS_K + row] = v.w;
            if (DUAL) {
                float4 u = *(const float4*)&W2[(long long)(kk + row) * Ncols + n0 + c * 4];
                w2t[(c * 4 + 0) * LDS_K + row] = u.x;
                w2t[(c * 4 + 1) * LDS_K + row] = u.y;
                w2t[(c * 4 + 2) * LDS_K + row] = u.z;
                w2t[(c * 4 + 3) * LDS_K + row] = u.w;
            }
        }
        __syncthreads();

        // --- 4 WMMA k-quads over this 16-wide chunk.
        //     A layout (§7.12.2): VGPR0 holds K={0,2}, VGPR1 K={1,3} per half.
#pragma unroll
        for (int ks = 0; ks < 16; ks += 4) {
            const int kq = ks + half * 2;
            v2f a1 = *(const v2f*)&xs[ml * LDS_K + kq];
            v2f a2;
            if (DUAL) a2 = *(const v2f*)&ns[ml * LDS_K + kq];
#pragma unroll
            for (int t = 0; t < 4; ++t) {
                const int noff = (cb + t * 16 + lidx) * LDS_K + kq;
                v2f b1 = *(const v2f*)&w1t[noff];
                acc[t] = __builtin_amdgcn_wmma_f32_16x16x4_f32(
                    false, a1, false, b1, (short)0, acc[t], false, false);
                if (DUAL) {
                    v2f b2 = *(const v2f*)&w2t[noff];
                    acc[t] = __builtin_amdgcn_wmma_f32_16x16x4_f32(
                        false, a2, false, b2, (short)0, acc[t], false, false);
                }
            }
        }
    }

    // --- epilogue: bias (+ ReLU), guarded store.  C/D layout (§7.12.2):
    //     VGPR i, lanes 0-15 -> M=i, lanes 16-31 -> M=i+8; N = lane&15.
    const int rbase = m0 + wm * 16 + half * 8;
#pragma unroll
    for (int t = 0; t < 4; ++t) {
        const int col = n0 + cb + t * 16 + lidx;
        const float bv = bias[col];
#pragma unroll
        for (int i = 0; i < 8; ++i) {
            int row = rbase + i;
            if (row < M) {
                float v = acc[t][i] + bv;
                if (RELU) v = fmaxf(v, 0.f);
                out[(long long)row * Ncols + col] = v;
            }
        }
    }
}

// ---------------------------------------------------------------------------
// Launch
// ---------------------------------------------------------------------------
extern "C" void kernel_launch(void* const* d_in, const int* in_sizes, int n_in,
                              void* d_out, int out_size, void* d_ws, size_t ws_size,
                              hipStream_t stream) {
    const float* x      = (const float*)d_in[0];
    const int*   src    = (const int*)d_in[1];
    const int*   dst    = (const int*)d_in[2];
    const float* W_self = (const float*)d_in[3];
    const float* W_neigh= (const float*)d_in[4];
    const float* b_enc  = (const float*)d_in[5];
    const float* W_dec  = (const float*)d_in[6];
    const float* b_dec  = (const float*)d_in[7];

    float* neigh = (float*)d_ws;                               // N*256
    float* deg   = neigh + (long long)N_NODES * IN_DIM;        // N

    float* recon = (float*)d_out;                              // N*256
    float* h     = recon + (long long)N_NODES * OUT_DIM;       // N*512

    // 1) zero accumulators (N*257 floats -> float4)
    {
        int n4 = (N_NODES * (IN_DIM + 1)) / 4;
        int blocks = (n4 + 255) / 256;
        zero_ws_kernel<<<blocks, 256, 0, stream>>>((float4*)neigh, n4);
    }
    // 2) edge scatter (64 threads/edge)
    {
        long long total = (long long)N_EDGES * 64;
        int blocks = (int)((total + 255) / 256);
        scatter_kernel<<<blocks, 256, 0, stream>>>(x, src, dst, neigh, deg);
    }
    // 3) mean normalize
    {
        int n4 = N_NODES * (IN_DIM / 4);
        int blocks = (n4 + 255) / 256;
        mean_kernel<<<blocks, 256, 0, stream>>>((float4*)neigh, deg, n4);
    }
    // 4) encoder: h = relu(x@W_self + neigh@W_neigh + b_enc)
    {
        dim3 grid((N_NODES + 63) / 64, HID_DIM / 128);
        gemm_wmma_kernel<true, true><<<grid, 256, 0, stream>>>(
            x, W_self, neigh, W_neigh, b_enc, h, N_NODES, IN_DIM, HID_DIM);
    }
    // 5) decoder: recon = h@W_dec + b_dec
    {
        dim3 grid((N_NODES + 63) / 64, OUT_DIM / 128);
        gemm_wmma_kernel<false, false><<<grid, 256, 0, stream>>>(
            h, W_dec, nullptr, nullptr, b_dec, recon, N_NODES, HID_DIM, OUT_DIM);
    }
}